// RoIHeads_29746943492598
// MI455X (gfx1250) — compile-verified
//
#include <hip/hip_runtime.h>

// ---------------- problem constants (match reference) ----------------
#define B_N   2
#define P_N   6000
#define C_N   81
#define CM1   80
#define M_N   (P_N * CM1)        // 480000 candidates per image
#define DET_N 100
#define SCORE_T 0.05f
#define NMS_IOU 0.5f
#define XCLIP 4.135166556742356f // log(1000/16)

#define NMS_T 768                // 24 wave32s per NMS block
#define VEC   4                  // floats per lane per staged chunk (b128)
#define CHUNK (NMS_T * VEC)      // 3072 elements per chunk
#define CAP   (M_N + CHUNK)      // compact capacity per image (+pad)

// compact workspace layout per image (floats):
//  S[CAP] | X1[CAP] | Y1[CAP] | X2[CAP] | Y2[CAP] | AR[CAP] | ORG(int)[CAP]
#define WS_IMG (7 * CAP)

#define DEC_BLOCKS 1500          // 256 thr = 8 wave32 = 8 proposals / block
#define BLK_PER_IMG 750

// ---------------------------------------------------------------------
__device__ __forceinline__ void wave_softmax(const float* lg, int lane,
                                             float& mx, float& inv) {
  float m = -INFINITY;
  for (int c = lane; c < C_N; c += 32) m = fmaxf(m, lg[c]);
  for (int d = 16; d; d >>= 1) m = fmaxf(m, __shfl_xor(m, d, 32));
  float s = 0.0f;
  for (int c = lane; c < C_N; c += 32) s += expf(lg[c] - m);
  for (int d = 16; d; d >>= 1) s += __shfl_xor(s, d, 32);
  mx = m; inv = 1.0f / s;
}

struct LaneDec { int valid; float score, x1, y1, x2, y2, area; };

__device__ __forceinline__ LaneDec lane_decode(
    const float* lg, const float* hrow, int c, int active,
    float mx, float inv, float pw, float ph, float pcx, float pcy,
    float fw, float fh) {
  LaneDec r;
  int cc = active ? c : 1;                     // keep loads in-bounds
  r.score = expf(lg[cc] - mx) * inv;
  const float* hc = hrow + (size_t)cc * 4;
  float dx = hc[0] / 10.0f, dy = hc[1] / 10.0f;
  float dw = fminf(hc[2] / 5.0f, XCLIP), dh = fminf(hc[3] / 5.0f, XCLIP);
  float cx = dx * pw + pcx, cy = dy * ph + pcy;
  float w = expf(dw) * pw, h = expf(dh) * ph;
  r.x1 = fminf(fmaxf(cx - 0.5f * w, 0.0f), fw);
  r.y1 = fminf(fmaxf(cy - 0.5f * h, 0.0f), fh);
  r.x2 = fminf(fmaxf(cx + 0.5f * w, 0.0f), fw);
  r.y2 = fminf(fmaxf(cy + 0.5f * h, 0.0f), fh);
  float bw = r.x2 - r.x1, bh = r.y2 - r.y1;
  r.area = bw * bh;                            // class offsets cancel in area
  r.valid = active && (r.score > SCORE_T) && (bw >= 0.01f) && (bh >= 0.01f);
  return r;
}

// ------------- kernel A: per-block live-candidate counts -------------
__global__ __launch_bounds__(256) void count_kernel(
    const float* __restrict__ logits, const float* __restrict__ hreg,
    const float* __restrict__ props,
    const int* __restrict__ pih, const int* __restrict__ piw,
    int* __restrict__ blkCnt) {
  int w = threadIdx.x >> 5, lane = threadIdx.x & 31;
  int gw = blockIdx.x * 8 + w;
  int img = gw / P_N, p = gw - img * P_N;
  float fh = (float)pih[0], fw = (float)piw[0];
  const float* pr = props + ((size_t)img * P_N + p) * 4;
  float pw = pr[2] - pr[0], ph = pr[3] - pr[1];
  float pcx = pr[0] + 0.5f * pw, pcy = pr[1] + 0.5f * ph;
  const float* lg = logits + ((size_t)img * P_N + p) * C_N;
  const float* hrow = hreg + ((size_t)img * P_N + p) * C_N * 4;
  float mx, inv; wave_softmax(lg, lane, mx, inv);
  int cnt = 0;
  for (int k = 0; k < 3; ++k) {
    int c = 1 + lane + 32 * k;
    LaneDec d = lane_decode(lg, hrow, c, c < C_N, mx, inv, pw, ph, pcx, pcy, fw, fh);
    cnt += __popc((unsigned)__ballot(d.valid));
  }
  __shared__ int wc[8];
  if (lane == 0) wc[w] = cnt;
  __syncthreads();
  if (threadIdx.x == 0) {
    int s = 0;
    for (int i = 0; i < 8; ++i) s += wc[i];
    blkCnt[blockIdx.x] = s;
  }
}

// ------------- kernel B: exclusive scan + -inf padding ---------------
__global__ __launch_bounds__(1024) void scan_kernel(
    const int* __restrict__ blkCnt, int* __restrict__ blkOff,
    int* __restrict__ MvArr, float* __restrict__ ws) {
  __shared__ int sMv[B_N];
  if (threadIdx.x == 0) {
    int b = 0;
    for (int img = 0; img < B_N; ++img) {
      int run = 0;
      for (int k = 0; k < BLK_PER_IMG; ++k, ++b) { blkOff[b] = run; run += blkCnt[b]; }
      MvArr[img] = run; sMv[img] = run;
    }
  }
  __syncthreads();
  for (int img = 0; img < B_N; ++img) {
    float* Sc = ws + (size_t)img * WS_IMG;
    int mv = sMv[img];
    for (int t = threadIdx.x; t < CHUNK; t += 1024) Sc[mv + t] = -INFINITY;
  }
}

// ------------- kernel C: stable compacting scatter -------------------
__global__ __launch_bounds__(256) void scatter_kernel(
    const float* __restrict__ logits, const float* __restrict__ hreg,
    const float* __restrict__ props,
    const int* __restrict__ pih, const int* __restrict__ piw,
    const int* __restrict__ blkOff, float* __restrict__ ws) {
  int w = threadIdx.x >> 5, lane = threadIdx.x & 31;
  int gw = blockIdx.x * 8 + w;
  int img = gw / P_N, p = gw - img * P_N;
  float fh = (float)pih[0], fw = (float)piw[0];
  float offmul = fmaxf(fh, fw) + 1.0f;
  const float* pr = props + ((size_t)img * P_N + p) * 4;
  float pw = pr[2] - pr[0], ph = pr[3] - pr[1];
  float pcx = pr[0] + 0.5f * pw, pcy = pr[1] + 0.5f * ph;
  const float* lg = logits + ((size_t)img * P_N + p) * C_N;
  const float* hrow = hreg + ((size_t)img * P_N + p) * C_N * 4;
  float mx, inv; wave_softmax(lg, lane, mx, inv);

  LaneDec d[3];
  int cnt = 0;
  for (int k = 0; k < 3; ++k) {
    int c = 1 + lane + 32 * k;
    d[k] = lane_decode(lg, hrow, c, c < C_N, mx, inv, pw, ph, pcx, pcy, fw, fh);
    cnt += __popc((unsigned)__ballot(d[k].valid));
  }
  __shared__ int wc[8];
  if (lane == 0) wc[w] = cnt;
  __syncthreads();
  int base = blkOff[blockIdx.x];
  for (int i = 0; i < w; ++i) base += wc[i];

  float* Sc = ws + (size_t)img * WS_IMG;
  float* X1 = Sc + (size_t)CAP;     float* Y1 = Sc + (size_t)2 * CAP;
  float* X2 = Sc + (size_t)3 * CAP; float* Y2 = Sc + (size_t)4 * CAP;
  float* AR = Sc + (size_t)5 * CAP; int* ORG = (int*)(Sc + (size_t)6 * CAP);

  for (int k = 0; k < 3; ++k) {
    int c = 1 + lane + 32 * k;
    unsigned m = (unsigned)__ballot(d[k].valid);
    if (d[k].valid) {
      int r = base + __popc(m & ((1u << lane) - 1u));
      float off = (float)c * offmul;
      Sc[r] = d[k].score;
      X1[r] = d[k].x1 + off; Y1[r] = d[k].y1 + off;
      X2[r] = d[k].x2 + off; Y2[r] = d[k].y2 + off;
      AR[r] = d[k].area;
      ORG[r] = p * CM1 + (c - 1);
    }
    base += __popc(m);
  }
}

// ------------- kernel D: fused greedy NMS over compact list ----------
// one block per image; per step: apply previous pick's suppression and
// compute argmax in ONE sweep; scores staged global->LDS with b128 async
// double buffering (ASYNCcnt); suppression stores ordered vs next sweep's
// async reloads with s_wait_storecnt 0 (counters are mutually unordered).
__global__ __launch_bounds__(NMS_T) void nms_kernel(
    float* __restrict__ ws, const int* __restrict__ MvArr,
    int* __restrict__ keep, int* __restrict__ kval) {
  const int img = blockIdx.x, tid = threadIdx.x;
  float* Sc = ws + (size_t)img * WS_IMG;
  const float* X1 = Sc + (size_t)CAP;     const float* Y1 = Sc + (size_t)2 * CAP;
  const float* X2 = Sc + (size_t)3 * CAP; const float* Y2 = Sc + (size_t)4 * CAP;
  const float* AR = Sc + (size_t)5 * CAP;
  const int* ORG = (const int*)(Sc + (size_t)6 * CAP);
  const int mv = MvArr[img];
  const int nch = (mv + CHUNK - 1) / CHUNK;

  __shared__ float redV[NMS_T / 32];
  __shared__ int   redI[NMS_T / 32];
  __shared__ float ch[5];                  // previous pick: x1,y1,x2,y2,area
  __shared__ float stage[2][CHUNK];
  unsigned lds0 = (unsigned)(size_t)&stage[0][tid * VEC];
  unsigned lds1 = (unsigned)(size_t)&stage[1][tid * VEC];

  for (int det = 0; det < DET_N; ++det) {
    float cx1 = ch[0], cy1 = ch[1], cx2 = ch[2], cy2 = ch[3], ca = ch[4];
    float bv = -INFINITY;
    int   bi = 0x7fffffff;
    if (nch > 0) {
      const float* g0 = Sc + (size_t)tid * VEC;
      asm volatile("global_load_async_to_lds_b128 %0, %1, off"
                   :: "v"(lds0), "v"(g0) : "memory");
      for (int chk = 0; chk < nch; ++chk) {
        if (chk + 1 < nch) {
          const float* gn = Sc + (size_t)(chk + 1) * CHUNK + (size_t)tid * VEC;
          unsigned dst = (chk & 1) ? lds0 : lds1;
          asm volatile("global_load_async_to_lds_b128 %0, %1, off"
                       :: "v"(dst), "v"(gn) : "memory");
          asm volatile("s_wait_asynccnt 1" ::: "memory");  // oldest chunk done
        } else {
          asm volatile("s_wait_asynccnt 0" ::: "memory");
        }
        const float* sb = &stage[chk & 1][tid * VEC];
        int j0 = chk * CHUNK + tid * VEC;
#pragma unroll
        for (int e = 0; e < VEC; ++e) {
          float v = sb[e];
          int j = j0 + e;
          if (det > 0 && v > -INFINITY) {   // self-suppression via IoU==1
            float iw = fmaxf(fminf(cx2, X2[j]) - fmaxf(cx1, X1[j]), 0.0f);
            float ih = fmaxf(fminf(cy2, Y2[j]) - fmaxf(cy1, Y1[j]), 0.0f);
            float inter = iw * ih;
            float iou = inter / fmaxf(AR[j] + ca - inter, 1e-8f);
            if (iou > NMS_IOU) { v = -INFINITY; Sc[j] = v; }
          }
          if (v > bv) { bv = v; bi = j; }   // strict > keeps first index
        }
      }
    }
    // wave32 reduction, first-index tie-break (matches jnp.argmax)
    for (int dd = 16; dd; dd >>= 1) {
      float ov = __shfl_xor(bv, dd, 32);
      int   oi = __shfl_xor(bi, dd, 32);
      if (ov > bv || (ov == bv && oi < bi)) { bv = ov; bi = oi; }
    }
    if ((tid & 31) == 0) { redV[tid >> 5] = bv; redI[tid >> 5] = bi; }
    __syncthreads();
    if (tid < 32) {
      float v = (tid < NMS_T / 32) ? redV[tid] : -INFINITY;
      int   i = (tid < NMS_T / 32) ? redI[tid] : 0x7fffffff;
      for (int dd = 16; dd; dd >>= 1) {
        float ov = __shfl_xor(v, dd, 32);
        int   oi = __shfl_xor(i, dd, 32);
        if (ov > v || (ov == v && oi < i)) { v = ov; i = oi; }
      }
      if (tid == 0) {
        if (i == 0x7fffffff) i = 0;
        int ok = (v > -INFINITY) ? 1 : 0;
        keep[img * DET_N + det] = ok ? ORG[i] : 0;
        kval[img * DET_N + det] = ok;
        ch[0] = X1[i]; ch[1] = Y1[i]; ch[2] = X2[i]; ch[3] = Y2[i]; ch[4] = AR[i];
      }
    }
    asm volatile("s_wait_storecnt 0" ::: "memory"); // order stores vs asyncs
    __syncthreads();
  }
}

// ------------- kernel E: gather + final decode of kept dets ----------
__global__ __launch_bounds__(256) void gather_kernel(
    const float* __restrict__ logits, const float* __restrict__ hreg,
    const float* __restrict__ oreg,  const float* __restrict__ props,
    const int* __restrict__ pih, const int* __restrict__ piw,
    const int* __restrict__ keep, const int* __restrict__ kval,
    float* __restrict__ out) {
  int t = blockIdx.x * blockDim.x + threadIdx.x;
  if (t >= B_N * DET_N) return;
  int img = t / DET_N, det = t - img * DET_N;
  float* o = out + (size_t)(img * DET_N + det) * 10;
  float* olab = out + (size_t)B_N * DET_N * 10 + (size_t)img * DET_N + det;
  int ok = kval[img * DET_N + det];
  if (!ok) {
    for (int k = 0; k < 10; ++k) o[k] = 0.0f;
    *olab = -1.0f;
    return;
  }
  float fh = (float)pih[0], fw = (float)piw[0];
  int idx = keep[img * DET_N + det];
  int p = idx / CM1;
  int c = idx - p * CM1 + 1;

  const float* pr = props + ((size_t)img * P_N + p) * 4;
  float pw = pr[2] - pr[0], ph = pr[3] - pr[1];
  float pcx = pr[0] + 0.5f * pw, pcy = pr[1] + 0.5f * ph;

  const float* lg = logits + ((size_t)img * P_N + p) * C_N;
  float mx = -INFINITY;
  for (int k = 0; k < C_N; ++k) mx = fmaxf(mx, lg[k]);
  float sm = 0.0f;
  for (int k = 0; k < C_N; ++k) sm += expf(lg[k] - mx);
  float score = expf(lg[c] - mx) / sm;

  const float* hc = hreg + ((size_t)img * P_N + p) * C_N * 4 + (size_t)c * 4;
  float dx = hc[0] / 10.0f, dy = hc[1] / 10.0f;
  float dw = fminf(hc[2] / 5.0f, XCLIP), dh = fminf(hc[3] / 5.0f, XCLIP);
  float cx = dx * pw + pcx, cy = dy * ph + pcy;
  float w = expf(dw) * pw, h = expf(dh) * ph;
  float x1 = fminf(fmaxf(cx - 0.5f * w, 0.0f), fw);
  float y1 = fminf(fmaxf(cy - 0.5f * h, 0.0f), fh);
  float x2 = fminf(fmaxf(cx + 0.5f * w, 0.0f), fw);
  float y2 = fminf(fmaxf(cy + 0.5f * h, 0.0f), fh);

  const float* oc = oreg + ((size_t)img * P_N + p) * C_N * 5 + (size_t)c * 5;
  float odx = oc[0] / 10.0f, ody = oc[1] / 10.0f;
  float odw = fminf(oc[2] / 5.0f, XCLIP), odh = fminf(oc[3] / 5.0f, XCLIP);
  float odt = oc[4] / 10.0f;
  float ocx = odx * pw + pcx, ocy = ody * ph + pcy;
  float ow = expf(odw) * pw, oh = expf(odh) * ph;

  o[0] = x1;  o[1] = y1;  o[2] = x2;  o[3] = y2;
  o[4] = ocx; o[5] = ocy; o[6] = ow;  o[7] = oh;
  o[8] = odt; o[9] = score;
  *olab = (float)c;
}

// ---------------------------------------------------------------------
extern "C" void kernel_launch(void* const* d_in, const int* in_sizes, int n_in,
                              void* d_out, int out_size, void* d_ws, size_t ws_size,
                              hipStream_t stream) {
  const float* logits = (const float*)d_in[0];
  const float* hreg   = (const float*)d_in[1];
  const float* oreg   = (const float*)d_in[2];
  const float* props  = (const float*)d_in[3];
  const int*   ih     = (const int*)d_in[4];
  const int*   iw     = (const int*)d_in[5];

  float* ws = (float*)d_ws;
  int* meta   = (int*)(ws + (size_t)2 * WS_IMG);
  int* blkCnt = meta;
  int* blkOff = meta + DEC_BLOCKS;
  int* MvArr  = meta + 2 * DEC_BLOCKS;
  int* keep   = MvArr + B_N;
  int* kval   = keep + B_N * DET_N;
  float* out  = (float*)d_out;

  count_kernel<<<DEC_BLOCKS, 256, 0, stream>>>(logits, hreg, props, ih, iw, blkCnt);
  scan_kernel<<<1, 1024, 0, stream>>>(blkCnt, blkOff, MvArr, ws);
  scatter_kernel<<<DEC_BLOCKS, 256, 0, stream>>>(logits, hreg, props, ih, iw, blkOff, ws);
  nms_kernel<<<B_N, NMS_T, 0, stream>>>(ws, MvArr, keep, kval);
  gather_kernel<<<1, 256, 0, stream>>>(logits, hreg, oreg, props, ih, iw,
                                       keep, kval, out);
}